// LIF_44143673868817
// MI455X (gfx1250) — compile-verified
//
#include <hip/hip_runtime.h>

// LIF spiking forward: [T=8, N=4194304] float32 recurrence over outermost T.
// Pure HBM-bandwidth problem (256 MiB @ 23.3 TB/s ~= 11.5 us). Strategy:
//  - one float4 of neurons per lane, u kept in VGPRs across all 8 timesteps
//  - gfx1250 async DMA (global_load_async_to_lds_b128) double-buffers the
//    next timestep's x-tile into LDS while the current one is consumed
//  - s_wait_asynccnt gates the pipeline (loads complete in order)
//  - non-temporal 128-bit stores for the write-once spike output

#define T_STEPS 8
#define BLOCK   256
#define VTH     1.0f
#define TAU     0.5f

typedef __attribute__((ext_vector_type(4))) float f4;

__device__ __forceinline__ void async_load_b128(unsigned lds_byte_addr,
                                                const void* gaddr) {
    // GLOBAL_LOAD_ASYNC_TO_LDS_B128, GV mode (64-bit vaddr, saddr=off).
    // VDST holds the per-lane LDS byte address; tracked by ASYNCcnt.
    asm volatile("global_load_async_to_lds_b128 %0, %1, off"
                 :
                 : "v"(lds_byte_addr), "v"(gaddr)
                 : "memory");
}

__global__ __launch_bounds__(BLOCK)
void lif_fwd_async(const float* __restrict__ x, float* __restrict__ o,
                   int stride_elems /* elements per timestep slice */) {
    // two ping-pong buffers: BLOCK lanes * 4 floats each
    __shared__ __align__(16) float smem[2 * BLOCK * 4];

    const int tid = threadIdx.x;
    const long long base = ((long long)blockIdx.x * BLOCK + tid) * 4;
    if (base >= (long long)stride_elems) return;

    float* s0p = &smem[tid * 4];
    float* s1p = &smem[BLOCK * 4 + tid * 4];
    // Flat-address truncation yields the LDS byte offset (aperture rule:
    // the low 32 bits of the flat address); matches what ds_load_b128 uses below.
    const unsigned s0 = (unsigned)(unsigned long long)s0p;
    const unsigned s1 = (unsigned)(unsigned long long)s1p;

    const float* xp = x + base;
    float*       op = o + base;

    // prime pipeline: stage t=0 into buffer 0
    async_load_b128(s0, (const void*)xp);

    f4 u = {0.0f, 0.0f, 0.0f, 0.0f};

#pragma unroll
    for (int t = 0; t < T_STEPS; ++t) {
        if (t + 1 < T_STEPS) {
            // prefetch next timestep into the other buffer
            async_load_b128((t & 1) ? s0 : s1,
                            (const void*)(xp + (long long)(t + 1) * stride_elems));
            // loads retire in order: <=1 outstanding means tile t has landed
            asm volatile("s_wait_asynccnt 1" ::: "memory");
        } else {
            asm volatile("s_wait_asynccnt 0" ::: "memory");
        }

        const float* scp = (t & 1) ? s1p : s0p;
        f4 xv = *(const f4*)scp;   // ds_load_b128 from this lane's slot

        // u_new = tau*u*(1-spike(u)) + x  ->  (u>VTH) ? x : fma(tau,u,x)
        f4 un;
        un.x = (u.x > VTH) ? xv.x : fmaf(TAU, u.x, xv.x);
        un.y = (u.y > VTH) ? xv.y : fmaf(TAU, u.y, xv.y);
        un.z = (u.z > VTH) ? xv.z : fmaf(TAU, u.z, xv.z);
        un.w = (u.w > VTH) ? xv.w : fmaf(TAU, u.w, xv.w);

        f4 ov;
        ov.x = (un.x > VTH) ? 1.0f : 0.0f;
        ov.y = (un.y > VTH) ? 1.0f : 0.0f;
        ov.z = (un.z > VTH) ? 1.0f : 0.0f;
        ov.w = (un.w > VTH) ? 1.0f : 0.0f;
        u = un;

        // spikes are write-once / never re-read: bypass L2 retention (NT)
        __builtin_nontemporal_store(ov, (f4*)(op + (long long)t * stride_elems));
    }
}

extern "C" void kernel_launch(void* const* d_in, const int* in_sizes, int n_in,
                              void* d_out, int out_size, void* d_ws, size_t ws_size,
                              hipStream_t stream) {
    const float* x = (const float*)d_in[0];
    float* o = (float*)d_out;

    const int total  = in_sizes[0];        // T * bs * C * H * W
    const int stride = total / T_STEPS;    // elements per timestep slice
    const int nvec   = (stride + 3) / 4;   // float4s per slice
    const int blocks = (nvec + BLOCK - 1) / BLOCK;

    lif_fwd_async<<<blocks, BLOCK, 0, stream>>>(x, o, stride);
}